// GlobalAttention_31275951849859
// MI455X (gfx1250) — compile-verified
//
#include <hip/hip_runtime.h>

#define TGT 64
#define SRC 64
#define BATCH 32
#define HID 512
#define ATT 512

typedef __attribute__((ext_vector_type(2))) float v2f;
typedef __attribute__((ext_vector_type(8))) float v8f;

// Hardware tanh on CDNA5 (v_tanh_f32, single TRANS op) with safe fallback.
__device__ __forceinline__ float fast_tanh(float x) {
#if defined(__has_builtin)
#if __has_builtin(__builtin_amdgcn_tanhf)
    return __builtin_amdgcn_tanhf(x);
#elif __has_builtin(__builtin_amdgcn_tanh_f32)
    return __builtin_amdgcn_tanh_f32(x);
#else
    return tanhf(x);
#endif
#else
    return tanhf(x);
#endif
}

// ---------------------------------------------------------------------------
// Kernel 1: fused projection GEMM via V_WMMA_F32_16X16X4_F32 (full fp32).
//   P[r][n], r in [0,4096), n in [0,512)
//   rows    0..2047 : h_part  (row = t*BATCH+b), X = h_t,  W = Wa[0:512]
//   rows 2048..4095 : s_part  (row = s*BATCH+b), X = src,  W = Wa[512:1024]
// One wave computes a 64x16 output strip (4 stacked 16x16 tiles) so each
// B fragment is reused by 4 WMMAs: per K=4 step -> 4 b64 A-loads + 2 b32
// B-loads feed 4 WMMA ops.
// ---------------------------------------------------------------------------
__global__ __launch_bounds__(128) void proj_gemm_kernel(
    const float* __restrict__ h_t,
    const float* __restrict__ src,
    const float* __restrict__ Wa,
    float* __restrict__ P)
{
    const int lane = threadIdx.x & 31;
    const int wave = threadIdx.x >> 5;
    const int tile = blockIdx.x * 4 + wave;       // 2048 strips of 64x16
    const int tilesN = ATT / 16;                  // 32
    const int tileM = tile / tilesN;              // 0..63 (64-row strips)
    const int tileN = tile % tilesN;

    const int rowBase = tileM * 64;               // multiple of 64, never straddles 2048
    const bool isH = rowBase < (TGT * BATCH);
    const float* __restrict__ X = isH ? (h_t + (size_t)rowBase * HID)
                                      : (src + (size_t)(rowBase - TGT * BATCH) * HID);
    const float* __restrict__ W = Wa + (size_t)(isH ? 0 : HID) * ATT + tileN * 16;

    // A 16x4 f32 layout: lanes 0-15 -> K=0,1 ; lanes 16-31 -> K=2,3 ; M = lane&15
    // B 4x16  f32 layout: lanes 0-15 -> K=0,1 ; lanes 16-31 -> K=2,3 ; N = lane&15
    const int m    = lane & 15;
    const int half = lane >> 4;

    v8f acc0 = {}, acc1 = {}, acc2 = {}, acc3 = {};
    for (int k0 = 0; k0 < HID; k0 += 4) {
        const int kA = k0 + half * 2;
        v2f b;
        b.x = W[(size_t)kA * ATT + m];
        b.y = W[(size_t)(kA + 1) * ATT + m];

        v2f a0 = *(const v2f*)(X + (size_t)(0 * 16 + m) * HID + kA);
        v2f a1 = *(const v2f*)(X + (size_t)(1 * 16 + m) * HID + kA);
        v2f a2 = *(const v2f*)(X + (size_t)(2 * 16 + m) * HID + kA);
        v2f a3 = *(const v2f*)(X + (size_t)(3 * 16 + m) * HID + kA);

        acc0 = __builtin_amdgcn_wmma_f32_16x16x4_f32(false, a0, false, b, (short)0, acc0, false, false);
        acc1 = __builtin_amdgcn_wmma_f32_16x16x4_f32(false, a1, false, b, (short)0, acc1, false, false);
        acc2 = __builtin_amdgcn_wmma_f32_16x16x4_f32(false, a2, false, b, (short)0, acc2, false, false);
        acc3 = __builtin_amdgcn_wmma_f32_16x16x4_f32(false, a3, false, b, (short)0, acc3, false, false);
    }

    // C/D layout: VGPR r -> row M = r + (lane>=16 ? 8 : 0), col N = lane&15
    float* __restrict__ out0 = P + (size_t)(rowBase + 0 * 16 + half * 8) * ATT + tileN * 16 + m;
    float* __restrict__ out1 = P + (size_t)(rowBase + 1 * 16 + half * 8) * ATT + tileN * 16 + m;
    float* __restrict__ out2 = P + (size_t)(rowBase + 2 * 16 + half * 8) * ATT + tileN * 16 + m;
    float* __restrict__ out3 = P + (size_t)(rowBase + 3 * 16 + half * 8) * ATT + tileN * 16 + m;
#pragma unroll
    for (int r = 0; r < 8; ++r) {
        out0[(size_t)r * ATT] = acc0[r];
        out1[(size_t)r * ATT] = acc1[r];
        out2[(size_t)r * ATT] = acc2[r];
        out3[(size_t)r * ATT] = acc3[r];
    }
}

// ---------------------------------------------------------------------------
// Kernel 2: fused  scores -> softmax -> context  per (t,b) pair.
// Never materializes the (64,64,32,512) "hidden" tensor (saves 268 MB of
// HBM round-trips). 256 threads = 8 waves; each wave owns 8 source rows;
// wave32 shuffle reductions; hardware v_tanh_f32 in the hot loop.
// ---------------------------------------------------------------------------
__global__ __launch_bounds__(256) void attn_fused_kernel(
    const float* __restrict__ src,
    const float* __restrict__ Va,
    const float* __restrict__ P,
    float* __restrict__ out)
{
    __shared__ float hp[ATT];
    __shared__ float va[ATT];
    __shared__ float attn[SRC];

    const int tb   = blockIdx.x;          // t*BATCH + b
    const int b    = tb & (BATCH - 1);
    const int tid  = threadIdx.x;
    const int lane = tid & 31;
    const int wave = tid >> 5;

    // Stage h_part row (t,b) and Va into LDS.
    const float* __restrict__ hpg = P + (size_t)tb * ATT;
    for (int i = tid; i < ATT; i += 256) {
        hp[i] = hpg[i];
        va[i] = Va[i];
    }
    __syncthreads();

    // scores[s] = sum_a tanh(hp[a] + sp[s][a]) * Va[a]   (sp rows hit in L2)
    for (int s = wave; s < SRC; s += 8) {
        const float* __restrict__ spg =
            P + (size_t)(TGT * BATCH + s * BATCH + b) * ATT;
        float acc = 0.f;
#pragma unroll
        for (int i = 0; i < ATT / 64; ++i) {          // 8 x float2 per lane
            const int a = i * 64 + lane * 2;
            const float2 sp2 = *(const float2*)(spg + a);
            const float x0 = hp[a] + sp2.x;
            const float x1 = hp[a + 1] + sp2.y;
            acc += fast_tanh(x0) * va[a] + fast_tanh(x1) * va[a + 1];
        }
#pragma unroll
        for (int off = 16; off; off >>= 1)
            acc += __shfl_xor(acc, off, 32);
        if (lane == 0) attn[s] = acc;
    }
    __syncthreads();

    // Softmax over s (64 values) by the first wave (2 values per lane).
    if (tid < 32) {
        float v0 = attn[lane], v1 = attn[lane + 32];
        float mx = fmaxf(v0, v1);
#pragma unroll
        for (int off = 16; off; off >>= 1)
            mx = fmaxf(mx, __shfl_xor(mx, off, 32));
        float e0 = __expf(v0 - mx), e1 = __expf(v1 - mx);
        float ssum = e0 + e1;
#pragma unroll
        for (int off = 16; off; off >>= 1)
            ssum += __shfl_xor(ssum, off, 32);
        float inv = 1.f / ssum;
        attn[lane]      = e0 * inv;
        attn[lane + 32] = e1 * inv;
    }
    __syncthreads();

    // context[t,b,h] = sum_s attn[s] * src[s,b,h]  (contiguous float2/thread)
    const int h0 = tid * 2;
    float c0 = 0.f, c1 = 0.f;
    for (int s = 0; s < SRC; ++s) {
        const float aw = attn[s];
        const float2 v = *(const float2*)(src + (size_t)(s * BATCH + b) * HID + h0);
        c0 += aw * v.x;
        c1 += aw * v.y;
    }
    float2 res; res.x = c0; res.y = c1;
    *(float2*)(out + (size_t)tb * HID + h0) = res;
}

// ---------------------------------------------------------------------------
extern "C" void kernel_launch(void* const* d_in, const int* in_sizes, int n_in,
                              void* d_out, int out_size, void* d_ws, size_t ws_size,
                              hipStream_t stream) {
    const float* h_t = (const float*)d_in[0];   // (64,32,512)
    const float* src = (const float*)d_in[1];   // (64,32,512)
    const float* Wa  = (const float*)d_in[2];   // (1024,512)
    const float* Va  = (const float*)d_in[3];   // (512,)
    float* out = (float*)d_out;                 // (64,32,512)

    // Workspace: P = [h_part ; s_part]  -> 4096*512 floats = 8 MB
    float* P = (float*)d_ws;

    // 2048 strips of 64x16, 4 waves/block -> 512 blocks
    proj_gemm_kernel<<<512, 128, 0, stream>>>(h_t, src, Wa, P);

    // One block per (t,b) pair
    attn_fused_kernel<<<TGT * BATCH, 256, 0, stream>>>(src, Va, P, out);
}